// ResidualConvBlock_19756849562053
// MI455X (gfx1250) — compile-verified
//
#include <hip/hip_runtime.h>

// ---------------------------------------------------------------------------
// Residual Chebyshev graph-conv block for MI455X (gfx1250, wave32, WMMA).
//   maps [B,N,F] f32, N=196608, B=4, F=16.
//   layer1: cheb(K=6, 8 nbrs) -> relu -> LN ; layer2: cheb(K=10, 20 nbrs)
//   -> relu -> LN ; out = x + maps.
// One kernel launch per Chebyshev step (global gather needs grid-wide sync).
// Dense F x F matmuls done with V_WMMA_F32_16X16X4_F32 (4 chained K=4 WMMAs
// per 16-pixel x 16-feature tile), accumulating across k through the C operand.
// ---------------------------------------------------------------------------

constexpr int   NPIXC   = 196608;
constexpr int   BATCHC  = 4;
constexpr int   FDIM    = 16;
constexpr int   TILES_PER_BATCH = NPIXC / 16;              // 12288
constexpr int   TOTAL_TILES     = BATCHC * TILES_PER_BATCH; // 49152
constexpr int   WPB     = 8;                                // waves per block
constexpr int   BLOCK   = 32 * WPB;                         // 256 threads
constexpr int   LDSROW  = 20;  // padded row stride (floats): conflict-free, 16B aligned
constexpr float EPS_LN  = 1e-5f;

typedef __attribute__((ext_vector_type(2))) float v2f;
typedef __attribute__((ext_vector_type(4))) float v4f;
typedef __attribute__((ext_vector_type(8))) float v8f;

// ---------------------------------------------------------------------------
// D(16x16) = A(16x16 from padded LDS tile) * B(16x16 = W[k] in LDS) + C
// via 4 chained V_WMMA_F32_16X16X4_F32.
// A layout (16x4 f32): lanes 0-15 rows M, VGPR0=K0 /K2(hi lanes), VGPR1=K1/K3.
// B layout (4x16):     VGPR0 = K-row {0|2}, VGPR1 = K-row {1|3}, col = lane&15.
// ---------------------------------------------------------------------------
__device__ __forceinline__ v8f tile_matmul_acc(const float* __restrict__ tileA,
                                               const float* __restrict__ Wsh,
                                               int lane, v8f c)
{
  const int row  = lane & 15;
  const int col  = lane & 15;
  const int koff = (lane >> 4) << 1;   // 0 for lanes 0-15, 2 for lanes 16-31
#pragma unroll
  for (int kk = 0; kk < 4; ++kk) {
    const int k0 = 4 * kk + koff;
    v2f a, b;
    a.x = tileA[row * LDSROW + k0];
    a.y = tileA[row * LDSROW + k0 + 1];
    b.x = Wsh[(k0    ) * FDIM + col];
    b.y = Wsh[(k0 + 1) * FDIM + col];
    c = __builtin_amdgcn_wmma_f32_16x16x4_f32(false, a, false, b,
                                              (short)0, c, false, false);
  }
  return c;
}

// ---------------------------------------------------------------------------
// acc = X @ W0  (k = 0 Chebyshev term, overwrites acc)
// ---------------------------------------------------------------------------
__global__ __launch_bounds__(BLOCK)
void mm0_kernel(const float* __restrict__ X, const float* __restrict__ W0,
                float* __restrict__ acc)
{
  __shared__ float sTile[WPB * 16 * LDSROW];
  __shared__ float sW[FDIM * FDIM];

  const int wave = threadIdx.x >> 5;
  const int lane = threadIdx.x & 31;
  const int tile = blockIdx.x * WPB + wave;
  const int b    = tile / TILES_PER_BATCH;
  const int p0   = (tile % TILES_PER_BATCH) * 16;
  const int row  = lane & 15;
  const int hi   = lane >> 4;
  const int f0   = hi * 8;

  sW[threadIdx.x] = W0[threadIdx.x];

  const size_t off = ((size_t)b * NPIXC + p0 + row) * FDIM + f0;
  const v4f x0 = *(const v4f*)(X + off);
  const v4f x1 = *(const v4f*)(X + off + 4);
  float* my = sTile + (wave * 16 + row) * LDSROW + f0;
  *(v4f*)(my)     = x0;
  *(v4f*)(my + 4) = x1;
  __syncthreads();

  v8f c = {};
  c = tile_matmul_acc(sTile + wave * 16 * LDSROW, sW, lane, c);

  const int    col     = lane & 15;
  const size_t accBase = ((size_t)b * NPIXC + p0) * FDIM;
#pragma unroll
  for (int v = 0; v < 8; ++v)
    acc[accBase + (size_t)(v + 8 * hi) * FDIM + col] = c[v];
}

// ---------------------------------------------------------------------------
// One Chebyshev step:
//   Tnew[p] = alpha * sum_j wl[p,j] * Tprev[idx[p,j]]  -  (Tpp ? Tpp[p] : 0)
//   acc    += Tnew_tile @ Wk          (WMMA, C loaded/stored from global acc)
// Tnew may alias Tpp (each thread reads its own Tpp element before writing;
// gathers only touch Tprev, a different buffer).
// ---------------------------------------------------------------------------
template <int NBR>
__global__ __launch_bounds__(BLOCK)
void cheb_step_kernel(const float* __restrict__ Tprev,
                      const float* Tpp,            // nullable / may alias Tnew
                      float*       Tnew,
                      float* __restrict__ acc,
                      const int*   __restrict__ idx,
                      const float* __restrict__ wl,
                      const float* __restrict__ Wk,
                      float alpha)
{
  __shared__ float sTile[WPB * 16 * LDSROW];
  __shared__ float sW[FDIM * FDIM];

  const int wave = threadIdx.x >> 5;
  const int lane = threadIdx.x & 31;
  const int tile = blockIdx.x * WPB + wave;
  const int b    = tile / TILES_PER_BATCH;
  const int p0   = (tile % TILES_PER_BATCH) * 16;
  const int row  = lane & 15;
  const int hi   = lane >> 4;
  const int f0   = hi * 8;
  const int p    = p0 + row;

  sW[threadIdx.x] = Wk[threadIdx.x];

  const float* Tb = Tprev + (size_t)b * NPIXC * FDIM;

  float r[8];
#pragma unroll
  for (int i = 0; i < 8; ++i) r[i] = 0.0f;

#pragma unroll
  for (int j = 0; j < NBR; ++j) {
    const int    nb = idx[p * NBR + j];
    const float  w  = wl[p * NBR + j];
    const float* s  = Tb + (size_t)nb * FDIM + f0;
    const v4f s0 = *(const v4f*)(s);
    const v4f s1 = *(const v4f*)(s + 4);
    r[0] = fmaf(w, s0.x, r[0]); r[1] = fmaf(w, s0.y, r[1]);
    r[2] = fmaf(w, s0.z, r[2]); r[3] = fmaf(w, s0.w, r[3]);
    r[4] = fmaf(w, s1.x, r[4]); r[5] = fmaf(w, s1.y, r[5]);
    r[6] = fmaf(w, s1.z, r[6]); r[7] = fmaf(w, s1.w, r[7]);
  }

  const size_t off = ((size_t)b * NPIXC + p) * FDIM + f0;
  if (Tpp != nullptr) {
    const v4f q0 = *(const v4f*)(Tpp + off);
    const v4f q1 = *(const v4f*)(Tpp + off + 4);
    r[0] = fmaf(alpha, r[0], -q0.x); r[1] = fmaf(alpha, r[1], -q0.y);
    r[2] = fmaf(alpha, r[2], -q0.z); r[3] = fmaf(alpha, r[3], -q0.w);
    r[4] = fmaf(alpha, r[4], -q1.x); r[5] = fmaf(alpha, r[5], -q1.y);
    r[6] = fmaf(alpha, r[6], -q1.z); r[7] = fmaf(alpha, r[7], -q1.w);
  } else {
#pragma unroll
    for (int i = 0; i < 8; ++i) r[i] *= alpha;
  }

  v4f o0 = { r[0], r[1], r[2], r[3] };
  v4f o1 = { r[4], r[5], r[6], r[7] };
  *(v4f*)(Tnew + off)     = o0;
  *(v4f*)(Tnew + off + 4) = o1;

  float* my = sTile + (wave * 16 + row) * LDSROW + f0;
  *(v4f*)(my)     = o0;
  *(v4f*)(my + 4) = o1;
  __syncthreads();

  const int    col     = lane & 15;
  const size_t accBase = ((size_t)b * NPIXC + p0) * FDIM;
  v8f c;
#pragma unroll
  for (int v = 0; v < 8; ++v)
    c[v] = acc[accBase + (size_t)(v + 8 * hi) * FDIM + col];

  c = tile_matmul_acc(sTile + wave * 16 * LDSROW, sW, lane, c);

#pragma unroll
  for (int v = 0; v < 8; ++v)
    acc[accBase + (size_t)(v + 8 * hi) * FDIM + col] = c[v];
}

// ---------------------------------------------------------------------------
// y = LayerNorm(relu(acc + bias)) * gamma + beta  [+ resid]   (per pixel, F=16)
// ---------------------------------------------------------------------------
__global__ __launch_bounds__(BLOCK)
void post_kernel(const float* __restrict__ acc,
                 const float* __restrict__ bias,
                 const float* __restrict__ gamma,
                 const float* __restrict__ beta,
                 const float* resid,               // nullable
                 float* __restrict__ out)
{
  const size_t pix = (size_t)blockIdx.x * blockDim.x + threadIdx.x;
  const float* a = acc + pix * FDIM;

  float v[16];
#pragma unroll
  for (int i = 0; i < 16; i += 4) {
    const v4f t = *(const v4f*)(a + i);
    v[i] = t.x; v[i + 1] = t.y; v[i + 2] = t.z; v[i + 3] = t.w;
  }

  float mu = 0.0f;
#pragma unroll
  for (int i = 0; i < 16; ++i) {
    v[i] = fmaxf(v[i] + bias[i], 0.0f);
    mu += v[i];
  }
  mu *= (1.0f / 16.0f);

  float var = 0.0f;
#pragma unroll
  for (int i = 0; i < 16; ++i) {
    const float d = v[i] - mu;
    var = fmaf(d, d, var);
  }
  var *= (1.0f / 16.0f);
  const float inv = rsqrtf(var + EPS_LN);

#pragma unroll
  for (int i = 0; i < 16; ++i)
    v[i] = fmaf((v[i] - mu) * inv, gamma[i], beta[i]);

  if (resid != nullptr) {
#pragma unroll
    for (int i = 0; i < 16; i += 4) {
      const v4f rr = *(const v4f*)(resid + pix * FDIM + i);
      v[i] += rr.x; v[i + 1] += rr.y; v[i + 2] += rr.z; v[i + 3] += rr.w;
    }
  }

#pragma unroll
  for (int i = 0; i < 16; i += 4) {
    v4f o = { v[i], v[i + 1], v[i + 2], v[i + 3] };
    *(v4f*)(out + pix * FDIM + i) = o;
  }
}

// ---------------------------------------------------------------------------
extern "C" void kernel_launch(void* const* d_in, const int* in_sizes, int n_in,
                              void* d_out, int out_size, void* d_ws, size_t ws_size,
                              hipStream_t stream)
{
  (void)in_sizes; (void)n_in; (void)out_size;

  const float* maps  = (const float*)d_in[0];
  const int*   idx1  = (const int*)  d_in[1];
  const float* wlap1 = (const float*)d_in[2];
  const int*   idx2  = (const int*)  d_in[3];
  const float* wlap2 = (const float*)d_in[4];
  const float* W1    = (const float*)d_in[5];
  const float* b1    = (const float*)d_in[6];
  const float* g1    = (const float*)d_in[7];
  const float* be1   = (const float*)d_in[8];
  const float* W2    = (const float*)d_in[9];
  const float* b2    = (const float*)d_in[10];
  const float* g2    = (const float*)d_in[11];
  const float* be2   = (const float*)d_in[12];
  float*       out   = (float*)d_out;

  const size_t MSZ = (size_t)BATCHC * NPIXC * FDIM;   // elements per map
  if (ws_size < 3 * MSZ * sizeof(float)) return;      // need acc, TA, TB
  float* acc = (float*)d_ws;
  float* TA  = acc + MSZ;
  float* TB  = TA + MSZ;

  const dim3 blk(BLOCK);
  const dim3 gTile(TOTAL_TILES / WPB);          // 6144 blocks, 1 wave = 1 tile
  const dim3 gPix((BATCHC * NPIXC) / BLOCK);    // 3072 blocks, 1 thr = 1 pixel

  // ------------------- layer 1: K=6, 8 neighbors -------------------
  mm0_kernel<<<gTile, blk, 0, stream>>>(maps, W1, acc);
  cheb_step_kernel<8><<<gTile, blk, 0, stream>>>(maps, nullptr, TA, acc,
                                                 idx1, wlap1, W1 + 1 * 256, 1.0f);
  cheb_step_kernel<8><<<gTile, blk, 0, stream>>>(TA, maps, TB, acc,
                                                 idx1, wlap1, W1 + 2 * 256, 2.0f);
  for (int k = 3; k < 6; ++k) {
    const float* prev = (k & 1) ? TB : TA;
    const float* pp   = (k & 1) ? TA : TB;
    float*       nw   = (k & 1) ? TA : TB;   // aliases pp (safe: own-pixel RMW)
    cheb_step_kernel<8><<<gTile, blk, 0, stream>>>(prev, pp, nw, acc,
                                                   idx1, wlap1, W1 + k * 256, 2.0f);
  }
  // layer-1 output -> TB (TA/TB dead after the last step for their cheb role)
  post_kernel<<<gPix, blk, 0, stream>>>(acc, b1, g1, be1, nullptr, TB);
  const float* X1 = TB;

  // ------------------- layer 2: K=10, 20 neighbors -----------------
  mm0_kernel<<<gTile, blk, 0, stream>>>(X1, W2, acc);
  cheb_step_kernel<20><<<gTile, blk, 0, stream>>>(X1, nullptr, TA, acc,
                                                  idx2, wlap2, W2 + 1 * 256, 1.0f);
  for (int k = 2; k < 10; ++k) {
    const float* prev = (k & 1) ? TB : TA;
    const float* pp   = (k & 1) ? TA : TB;   // k=2: pp = TB = X1 (last use)
    float*       nw   = (k & 1) ? TA : TB;   // aliases pp (safe: own-pixel RMW)
    cheb_step_kernel<20><<<gTile, blk, 0, stream>>>(prev, pp, nw, acc,
                                                    idx2, wlap2, W2 + k * 256, 2.0f);
  }
  post_kernel<<<gPix, blk, 0, stream>>>(acc, b2, g2, be2, maps, out);
}